// CIRNet_69166153335527
// MI455X (gfx1250) — compile-verified
//
#include <hip/hip_runtime.h>
#include <math.h>

typedef __attribute__((ext_vector_type(2))) float v2f;
typedef __attribute__((ext_vector_type(8))) float v8f;

#define FEAT 18
#define SNUM 8

// ---------------------------------------------------------------------------
// Kernel 1: per-row parallel part. One wave handles a tile of 16 rows.
// sig/eps dot products done with 4 chained V_WMMA_F32_16X16X4_F32 (exact f32).
//   A (16x16, as 4 slices of 16x4): A[M][f] = trace[i0+M][2+f]
//   B (16x16, as 4 slices of 4x16): col0 = sigma_W (f<8), col1 = eps_W (f>=8)
//   D col0 = raw sig, D col1 = eps.
// Writes: out[N..2N) = regs, out[2N..3N) = dts, ws dc[i] = {dt, sig*eps*sqrt|dt|}
// ---------------------------------------------------------------------------
__global__ __launch_bounds__(256) void cir_coeff_kernel(
    const float* __restrict__ trace,
    const float* __restrict__ sW,    // 8
    const float* __restrict__ sb,    // 1
    const float* __restrict__ eW,    // 8
    const float* __restrict__ kp,    // 1
    const float* __restrict__ thp,   // 1
    float* __restrict__ out,         // 3*N floats (r, regs, dts)
    float2* __restrict__ dc,         // N {dt, c} records
    int N)
{
    __shared__ float sigS[8][16];
    __shared__ float epsS[8][16];

    const int lane = threadIdx.x & 31;
    const int wv   = threadIdx.x >> 5;
    const int ntiles = (N + 15) >> 4;

    // Clamp tile (never branch around the WMMAs -> EXEC stays all-ones).
    int tile = blockIdx.x * 8 + wv;
    if (tile > ntiles - 1) tile = ntiles - 1;
    const int i0 = tile << 4;

    const int n  = lane & 15;   // A: row-in-tile (M); B/D: column (N)
    const int hi = lane >> 4;   // 0: K in {0,1}; 1: K in {2,3}

    // A fragment source pointer for this lane's (clamped) row.
    int row = i0 + n; if (row > N - 1) row = N - 1;
    const float* rp = trace + (long)row * FEAT + 2;
    const int ko = hi ? 2 : 0;

    v8f acc = {0.f, 0.f, 0.f, 0.f, 0.f, 0.f, 0.f, 0.f};

#if defined(__AMDGCN__)
    #pragma unroll
    for (int kb = 0; kb < 4; ++kb) {
        v2f a;
        a.x = rp[4 * kb + ko];
        a.y = rp[4 * kb + ko + 1];

        // B fragment: lane<16 holds K rows {0,1}; lane>=16 holds {2,3}.
        const int fx = 4 * kb + (hi ? 2 : 0);
        const int fy = 4 * kb + (hi ? 3 : 1);
        // Only columns 0 (sigma) and 1 (eps) are non-zero. Ternaries only
        // (selects, no EXEC manipulation before the WMMA).
        float bx = (n == 0) ? ((fx < SNUM) ? sW[fx] : 0.f)
                 : (n == 1) ? ((fx >= SNUM) ? eW[fx - SNUM] : 0.f) : 0.f;
        float by = (n == 0) ? ((fy < SNUM) ? sW[fy] : 0.f)
                 : (n == 1) ? ((fy >= SNUM) ? eW[fy - SNUM] : 0.f) : 0.f;
        v2f b; b.x = bx; b.y = by;

        acc = __builtin_amdgcn_wmma_f32_16x16x4_f32(
            /*neg_a=*/false, a, /*neg_b=*/false, b,
            /*c_mod=*/(short)0, acc, /*reuse_a=*/false, /*reuse_b=*/false);
    }
#endif

    // D layout: element (M, Ncol): lane Ncol holds M=0..7 in acc[0..7],
    // lane Ncol+16 holds M=8..15. We need columns 0 (sig) and 1 (eps).
    if (lane == 0) {
        #pragma unroll
        for (int m = 0; m < 8; ++m) sigS[wv][m] = acc[m];
    } else if (lane == 16) {
        #pragma unroll
        for (int m = 0; m < 8; ++m) sigS[wv][8 + m] = acc[m];
    } else if (lane == 1) {
        #pragma unroll
        for (int m = 0; m < 8; ++m) epsS[wv][m] = acc[m];
    } else if (lane == 17) {
        #pragma unroll
        for (int m = 0; m < 8; ++m) epsS[wv][8 + m] = acc[m];
    }
    __syncthreads();

    // Finalize: 16 lanes, one row each. (Clamped duplicate tiles rewrite the
    // last tile with identical values -- benign.)
    const int i = i0 + n;
    if (hi == 0 && i < N) {
        const float sraw = sigS[wv][n];
        const float e    = epsS[wv][n];
        const float s    = sraw + sb[0];
        const float t0v  = trace[(long)i * FEAT];
        const float t1v  = trace[(long)(i + 1) * FEAT];
        const float dt   = t1v - t0v;
        const float kk   = kp[0];
        const float th   = thp[0];
        out[2 * N + i] = dt;                       // dts
        out[N + i]     = 2.0f * kk * th - s * s;   // regs
        const float c  = s * e * sqrtf(fabsf(dt)); // sqrt|r*dt| = sqrt|r|*sqrt|dt|
        dc[i] = make_float2(dt, c);
    }
}

// ---------------------------------------------------------------------------
// Kernel 2: the inherently serial nonlinear scan, single lane.
//   r <- r + k*(th-r)*dt + c*sqrt(|r|)
// Double-buffered 32-deep register blocks of {dt,c} so the (L2-resident)
// loads overlap the dependent FP chain.
// ---------------------------------------------------------------------------
__global__ void cir_scan_kernel(
    const float* __restrict__ trace,
    const float* __restrict__ kp,
    const float* __restrict__ thp,
    const float2* __restrict__ dc,
    float* __restrict__ out_r,
    int N)
{
    if (threadIdx.x != 0 || blockIdx.x != 0) return;

    const float kk = kp[0];
    const float th = thp[0];
    float r = trace[1];          // trace[0, 1]

    constexpr int U = 32;
    float2 A[U], B[U];
    const int Nfull = N & ~(2 * U - 1);

    // Preload block A = [0, U)
    #pragma unroll
    for (int u = 0; u < U; ++u) {
        int j = u; if (j > N - 1) j = N - 1;
        A[u] = dc[j];
    }

    for (int base = 0; base < Nfull; base += 2 * U) {
        const int nb = base + U;
        #pragma unroll
        for (int u = 0; u < U; ++u) {          // load B while computing A
            int j = nb + u; if (j > N - 1) j = N - 1;
            B[u] = dc[j];
        }
        #pragma unroll
        for (int u = 0; u < U; ++u) {
            const float dt = A[u].x, c = A[u].y;
            const float sq = sqrtf(fabsf(r));
            r = fmaf(kk * dt, th - r, r) + c * sq;
            out_r[base + u] = r;
        }
        const int nb2 = base + 2 * U;
        #pragma unroll
        for (int u = 0; u < U; ++u) {          // reload A while computing B
            int j = nb2 + u; if (j > N - 1) j = N - 1;
            A[u] = dc[j];
        }
        #pragma unroll
        for (int u = 0; u < U; ++u) {
            const float dt = B[u].x, c = B[u].y;
            const float sq = sqrtf(fabsf(r));
            r = fmaf(kk * dt, th - r, r) + c * sq;
            out_r[nb + u] = r;
        }
    }
    for (int i = Nfull; i < N; ++i) {
        const float2 v = dc[i];
        const float sq = sqrtf(fabsf(r));
        r = fmaf(kk * v.x, th - r, r) + v.y * sq;
        out_r[i] = r;
    }
}

// ---------------------------------------------------------------------------
extern "C" void kernel_launch(void* const* d_in, const int* in_sizes, int n_in,
                              void* d_out, int out_size, void* d_ws, size_t ws_size,
                              hipStream_t stream) {
    const float* trace = (const float*)d_in[0];
    const float* sW    = (const float*)d_in[1];
    const float* sb    = (const float*)d_in[2];
    const float* eW    = (const float*)d_in[3];
    const float* kp    = (const float*)d_in[4];
    const float* thp   = (const float*)d_in[5];
    float* out = (float*)d_out;

    const int T = in_sizes[0] / FEAT;
    const int N = T - 1;

    float2* dc = (float2*)d_ws;  // needs 8*N bytes (~8.4 MB)

    const int ntiles = (N + 15) / 16;
    const int blocks = (ntiles + 7) / 8;   // 8 waves (tiles) per 256-thread block

    cir_coeff_kernel<<<blocks, 256, 0, stream>>>(trace, sW, sb, eW, kp, thp,
                                                 out, dc, N);
    cir_scan_kernel<<<1, 32, 0, stream>>>(trace, kp, thp, dc, out, N);
}